// VecAdder_29875792511508
// MI455X (gfx1250) — compile-verified
//
#include <hip/hip_runtime.h>

// Spiking ripple-carry adder == one 32-bit integer add per row.
// A, B: [N,32] float32 in {0.0,1.0}. Out: S [N,32] floats then Cout [N].
//
// Memory-bound kernel (388 B of HBM traffic per row; ~16.6 us floor at
// 23.3 TB/s). CDNA5 streaming shape: inputs are DMA'd global->LDS with
// GLOBAL_LOAD_ASYNC_TO_LDS_B128 (ASYNCcnt path, nontemporal so 256 MB of
// single-use data doesn't rinse the 192 MB L2), synchronized with
// s_wait_asynccnt + workgroup barrier, packed from LDS via ds_load_b128,
// and written back as nontemporal b128 stores.

typedef unsigned int       u32;
typedef unsigned long long u64;
typedef __attribute__((ext_vector_type(4))) unsigned int v4u;
typedef __attribute__((ext_vector_type(4))) float        v4f;

#define ROW_BITS        32
#define ROW_BYTES       128            // 32 floats
#define ROWS_PER_BLOCK  128            // == blockDim.x, one row per thread

// INST_OFFSET is added to BOTH the LDS dest address and the global src
// address (CDNA5 ISA), so one base pair walks the row.
#define ASYNC_LD_B128(ldsoff, gaddr, IMM)                                  \
    asm volatile("global_load_async_to_lds_b128 %0, %1, off offset:" #IMM  \
                 " th:TH_LOAD_NT"                                          \
                 :: "v"(ldsoff), "v"(gaddr) : "memory")

__global__ __launch_bounds__(ROWS_PER_BLOCK) void rc_adder32_async_kernel(
    const u32* __restrict__ A,    // raw float bits of [N,32]
    const u32* __restrict__ B,    // raw float bits of [N,32]
    float* __restrict__ S,        // [N,32]
    float* __restrict__ Cout,     // [N]
    int nrows)
{
    // A-tile then B-tile: 2 * 128 rows * 128 B = 32 KB (of 320 KB/WGP).
    __shared__ __align__(16) unsigned char smem[2 * ROWS_PER_BLOCK * ROW_BYTES];

    const int tid = threadIdx.x;
    int row = blockIdx.x * ROWS_PER_BLOCK + tid;
    if (row >= nrows) row = nrows - 1;   // clamp: keep block convergent

    unsigned char* aRow = smem + (size_t)tid * ROW_BYTES;
    unsigned char* bRow = smem + (size_t)(ROWS_PER_BLOCK + tid) * ROW_BYTES;

    // Workgroup-relative LDS byte offsets for the async instruction's VDST.
    typedef __attribute__((address_space(3))) unsigned char lds_byte;
    unsigned aoff = (unsigned)(u64)(lds_byte*)aRow;
    unsigned boff = (unsigned)(u64)(lds_byte*)bRow;

    u64 ga = (u64)(const void*)(A + (size_t)row * ROW_BITS);
    u64 gb = (u64)(const void*)(B + (size_t)row * ROW_BITS);

    // 8x b128 per operand, global -> LDS, tracked on ASYNCcnt.
    ASYNC_LD_B128(aoff, ga, 0);
    ASYNC_LD_B128(aoff, ga, 16);
    ASYNC_LD_B128(aoff, ga, 32);
    ASYNC_LD_B128(aoff, ga, 48);
    ASYNC_LD_B128(aoff, ga, 64);
    ASYNC_LD_B128(aoff, ga, 80);
    ASYNC_LD_B128(aoff, ga, 96);
    ASYNC_LD_B128(aoff, ga, 112);
    ASYNC_LD_B128(boff, gb, 0);
    ASYNC_LD_B128(boff, gb, 16);
    ASYNC_LD_B128(boff, gb, 32);
    ASYNC_LD_B128(boff, gb, 48);
    ASYNC_LD_B128(boff, gb, 64);
    ASYNC_LD_B128(boff, gb, 80);
    ASYNC_LD_B128(boff, gb, 96);
    ASYNC_LD_B128(boff, gb, 112);

    // Own wave's async ops complete, then all waves' (barrier).
    asm volatile("s_wait_asynccnt 0x0" ::: "memory");
    __syncthreads();

    // Pack each row into a uint32 straight out of LDS (ds_load_b128).
    // For {0.0f, 1.0f}, bit 23 (exponent LSB) of the raw float IS the bit.
    const v4u* a4 = (const v4u*)aRow;
    const v4u* b4 = (const v4u*)bRow;

    u32 a = 0, b = 0;
#pragma unroll
    for (int k = 0; k < 8; ++k) {
        v4u va = a4[k];
        v4u vb = b4[k];
        a |= ((va.x >> 23) & 1u) << (4 * k + 0);
        a |= ((va.y >> 23) & 1u) << (4 * k + 1);
        a |= ((va.z >> 23) & 1u) << (4 * k + 2);
        a |= ((va.w >> 23) & 1u) << (4 * k + 3);
        b |= ((vb.x >> 23) & 1u) << (4 * k + 0);
        b |= ((vb.y >> 23) & 1u) << (4 * k + 1);
        b |= ((vb.z >> 23) & 1u) << (4 * k + 2);
        b |= ((vb.w >> 23) & 1u) << (4 * k + 3);
    }

    // The entire gate network: one add + carry-out.
    u32 s     = a + b;
    u32 carry = (s < a) ? 1u : 0u;

    // Unpack sum bits to floats; nontemporal b128 stores.
    v4f* s4 = (v4f*)(S + (size_t)row * ROW_BITS);
#pragma unroll
    for (int k = 0; k < 8; ++k) {
        v4f v;
        v.x = (float)((s >> (4 * k + 0)) & 1u);
        v.y = (float)((s >> (4 * k + 1)) & 1u);
        v.z = (float)((s >> (4 * k + 2)) & 1u);
        v.w = (float)((s >> (4 * k + 3)) & 1u);
        __builtin_nontemporal_store(v, s4 + k);
    }

    // Coalesced: thread r writes Cout[r].
    Cout[row] = (float)carry;
}

extern "C" void kernel_launch(void* const* d_in, const int* in_sizes, int n_in,
                              void* d_out, int out_size, void* d_ws, size_t ws_size,
                              hipStream_t stream) {
    (void)n_in; (void)d_ws; (void)ws_size; (void)out_size;

    const u32* A = (const u32*)d_in[0];   // [N,32] float32 bits
    const u32* B = (const u32*)d_in[1];   // [N,32] float32 bits

    int nrows = in_sizes[0] / ROW_BITS;   // N = 1048576

    float* S    = (float*)d_out;                             // N*32 floats
    float* Cout = (float*)d_out + (size_t)nrows * ROW_BITS;  // then N floats

    int blocks = (nrows + ROWS_PER_BLOCK - 1) / ROWS_PER_BLOCK;
    rc_adder32_async_kernel<<<blocks, ROWS_PER_BLOCK, 0, stream>>>(A, B, S, Cout, nrows);
}